// PaiIndexMatrix_10934986736323
// MI455X (gfx1250) — compile-verified
//
#include <hip/hip_runtime.h>
#include <math.h>

typedef __attribute__((ext_vector_type(2))) float v2f;
typedef __attribute__((ext_vector_type(8))) float v8f;

constexpr int B_ = 8;
constexpr int N_ = 4096;
constexpr int K_ = 20;
constexpr int M_ = 32;
constexpr int LDS_STRIDE = 260;   // 256 cols + pad to avoid bank conflicts

// order-preserving float -> uint map (larger float => larger uint)
__device__ __forceinline__ unsigned fkey(float v) {
    unsigned ub = __float_as_uint(v);
    ub ^= (unsigned)((int)ub >> 31) | 0x80000000u;
    return ub;
}

// ---------------------------------------------------------------------------
// Kernel 1: per 16-row tile, compute 16x4096 score block via WMMA f32 16x16x4
// (score = 2*xi.xj - xx_j, same per-row ordering as -||xi-xj||^2), then
// per-row top-20 selection (descending, lowest-index tie-break) -> indices.
// Workgroup = 16 waves (512 threads, wave32). Wave w: WMMA col-block w of
// each 256-col chunk, then selection for row w. Selection state is a per-lane
// sorted list of packed 64-bit keys: {monotonic(value), ~index} so all
// comparisons are single v_cmp_gt_u64 + cndmask (no branches).
// ---------------------------------------------------------------------------
__global__ __launch_bounds__(512)
void knn_topk_kernel(const float* __restrict__ x, float* __restrict__ outIdx) {
    __shared__ float sc[16 * LDS_STRIDE];

    const int wg   = blockIdx.x;
    const int b    = wg >> 8;            // 256 row-tiles per batch
    const int r0   = (wg & 255) << 4;    // row base within batch
    const int tid  = threadIdx.x;
    const int wave = tid >> 5;
    const int lane = tid & 31;
    const int half = lane >> 4;
    const int l15  = lane & 15;

    const float* xb = x + (size_t)b * 3 * N_;

    // ---- A fragment (identical in every wave): row m = l15 ----
    // A[m] = (2*x0, 2*x1, 2*x2, -1); lanes 0-15 hold K=(0,1), lanes 16-31 K=(2,3)
    const int mrow = r0 + l15;
    const float ax0 = xb[0 * N_ + mrow];
    const float ax1 = xb[1 * N_ + mrow];
    const float ax2 = xb[2 * N_ + mrow];
    v2f afrag;
    afrag.x = (half == 0) ? (2.0f * ax0) : (2.0f * ax2);
    afrag.y = (half == 0) ? (2.0f * ax1) : -1.0f;

    // ---- per-lane sorted top-K key list (descending; 0 = empty sentinel) ----
    unsigned long long keys[K_];
#pragma unroll
    for (int s = 0; s < K_; ++s) keys[s] = 0ull;

    // ---- stream 4096 columns in 16 chunks of 256 ----
#pragma unroll 1
    for (int chunk = 0; chunk < 16; ++chunk) {
        // B fragment: col j; B col = (x0, x1, x2, xx)
        const int j  = chunk * 256 + wave * 16 + l15;
        const float b0 = xb[0 * N_ + j];
        const float b1 = xb[1 * N_ + j];
        const float b2 = xb[2 * N_ + j];
        const float bxx = b0 * b0 + b1 * b1 + b2 * b2;
        v2f bfrag;
        bfrag.x = (half == 0) ? b0 : b2;
        bfrag.y = (half == 0) ? b1 : bxx;

        v8f acc = {};
        // D(16x16) = A(16x4) x B(4x16): s_mj = 2*x_m.x_j - xx_j
        acc = __builtin_amdgcn_wmma_f32_16x16x4_f32(
            false, afrag, false, bfrag, (short)0, acc, false, false);

        __syncthreads();   // previous chunk's LDS reads are done
#pragma unroll
        for (int e = 0; e < 8; ++e) {
            // D layout: vgpr e -> row (half*8 + e), col l15 (block-local)
            sc[(half * 8 + e) * LDS_STRIDE + wave * 16 + l15] = acc[e];
        }
        __syncthreads();   // chunk scores visible to all waves

        // ---- selection scan: wave w owns row w ----
        const int base = chunk * 256;
#pragma unroll
        for (int t = 0; t < 8; ++t) {
            const int   col = lane + t * 32;
            const float v   = sc[wave * LDS_STRIDE + col];
            const int   ci  = base + col;
            unsigned long long ck =
                ((unsigned long long)fkey(v) << 32) | (unsigned)(~ci);
            // wave-uniform fast reject (scalar branch, single u64 compare)
            if (__ballot(ck > keys[K_ - 1]) != 0ull) {
                // flat sorted insertion: 1 cmp + 4 cndmask per slot
#pragma unroll
                for (int s = 0; s < K_; ++s) {
                    const bool better = ck > keys[s];
                    const unsigned long long mx = better ? ck : keys[s];
                    const unsigned long long mn = better ? keys[s] : ck;
                    keys[s] = mx;
                    ck = mn;
                }
            }
        }
    }

    // ---- merge 32 per-lane lists: pop global max K_ times (descending) ----
    const long long pbase = ((long long)(b * N_ + r0 + wave)) * K_;
#pragma unroll 1
    for (int k = 0; k < K_; ++k) {
        unsigned long long best = keys[0];
#pragma unroll
        for (int off = 16; off >= 1; off >>= 1) {
            const unsigned long long o = __shfl_xor(best, off, 32);
            best = (o > best) ? o : best;
        }
        const int ii = (int)(~(unsigned)best);   // column index in [0, N)
        if (lane == 0) outIdx[pbase + k] = (float)(b * N_ + ii);
        // unique winner lane (keys are globally unique) shifts its list
        const bool win = (keys[0] == best);
#pragma unroll
        for (int s = 0; s < K_ - 1; ++s) keys[s] = win ? keys[s + 1] : keys[s];
        keys[K_ - 1] = win ? 0ull : keys[K_ - 1];
    }
}

// ---------------------------------------------------------------------------
// Kernel 2: per point, gather 20 neighbor coords, x_rel = coord - coord[0],
// permatrix = relu(x_rel @ kernels + one_padding), then topkmax over K.
// One wave per point; lane = kernel column m (M=32 == wave32 width).
// ---------------------------------------------------------------------------
__global__ __launch_bounds__(256)
void permatrix_kernel(const float* __restrict__ x,
                      const float* __restrict__ kernels,
                      const float* __restrict__ one_padding,
                      const float* __restrict__ outIdx,
                      float* __restrict__ outP) {
    const int wave = threadIdx.x >> 5;
    const int lane = threadIdx.x & 31;
    const int p    = blockIdx.x * 8 + wave;      // 0 .. B*N-1
    const int b    = p >> 12;                    // N = 4096
    const float* xb = x + (size_t)b * 3 * N_;

    // lanes 0..19 load neighbor-k coordinates
    float cx = 0.f, cy = 0.f, cz = 0.f;
    if (lane < K_) {
        const int g  = (int)outIdx[(long long)p * K_ + lane];
        const int nl = g - b * N_;
        cx = xb[0 * N_ + nl];
        cy = xb[1 * N_ + nl];
        cz = xb[2 * N_ + nl];
    }
    const float r0x = __shfl(cx, 0, 32);
    const float r0y = __shfl(cy, 0, 32);
    const float r0z = __shfl(cz, 0, 32);

    const float k0 = kernels[0 * M_ + lane];
    const float k1 = kernels[1 * M_ + lane];
    const float k2 = kernels[2 * M_ + lane];

    float vals[K_];
    float s1 = 0.f;
#pragma unroll
    for (int k = 0; k < K_; ++k) {
        const float dx = __shfl(cx, k, 32) - r0x;
        const float dy = __shfl(cy, k, 32) - r0y;
        const float dz = __shfl(cz, k, 32) - r0z;
        float v = dx * k0 + dy * k1 + dz * k2 + one_padding[k * M_ + lane];
        v = fmaxf(v, 0.f);
        vals[k] = v;
        s1 += v;
    }
    const float inv1 = 1.0f / (s1 + 1e-6f);
    float s2 = 0.f;
#pragma unroll
    for (int k = 0; k < K_; ++k) {
        float t = vals[k] * inv1;
        t = t * t;
        vals[k] = t;
        s2 += t;
    }
    const float inv2 = 1.0f / (s2 + 1e-6f);
    float* o = outP + (long long)p * K_ * M_;
#pragma unroll
    for (int k = 0; k < K_; ++k) {
        const float t = vals[k] * inv2;
        o[k * M_ + lane] = (t > 0.1f) ? t : 0.f;
    }
}

// ---------------------------------------------------------------------------
extern "C" void kernel_launch(void* const* d_in, const int* in_sizes, int n_in,
                              void* d_out, int out_size, void* d_ws, size_t ws_size,
                              hipStream_t stream) {
    const float* x           = (const float*)d_in[0];   // (B,3,N) f32
    const float* kernels     = (const float*)d_in[1];   // (3,M)  f32
    const float* one_padding = (const float*)d_in[2];   // (K,M)  f32

    float* out    = (float*)d_out;
    float* outIdx = out;                                // B*N*K  (indices as float)
    float* outP   = out + (size_t)B_ * N_ * K_;         // B*N*K*M

    // 16-row tiles: B*N/16 = 2048 workgroups of 16 waves
    knn_topk_kernel<<<dim3(B_ * N_ / 16), dim3(512), 0, stream>>>(x, outIdx);
    // one wave per point: 8 waves/block -> B*N/8 = 4096 blocks
    permatrix_kernel<<<dim3(B_ * N_ / 8), dim3(256), 0, stream>>>(
        x, kernels, one_padding, outIdx, outP);
}